// GNNLocalCluster_6158983102549
// MI455X (gfx1250) — compile-verified
//
#include <hip/hip_runtime.h>
#include <math.h>

// ---- CDNA5 (gfx1250) fully-fused GNN local-cluster kernel ----
// One workgroup (256 threads = 8 wave32) per batch-window b in [0,784).
// All intermediates live in LDS; matrix math uses V_WMMA_F32_16X16X4_F32;
// input tile staging uses GLOBAL_LOAD_ASYNC_TO_LDS_B128 (ASYNCcnt path).

typedef float v2f __attribute__((ext_vector_type(2)));
typedef float v8f __attribute__((ext_vector_type(8)));
typedef int   v4i __attribute__((vector_size(16)));   // matches builtin param type

#define C_IN    64
#define CF      16
#define NN      256      // nodes per window (16x16 grid)
#define KSEL    9
#define XF_STR  17       // padded row stride (coprime with 64 banks)
#define SIM_STR 273      // padded sim row stride
#define HW      12544    // 112*112

#if defined(__has_builtin)
#if __has_builtin(__builtin_amdgcn_global_load_async_to_lds_b128)
#define HAVE_ASYNC_LDS_BUILTIN 1
#endif
#if __has_builtin(__builtin_amdgcn_s_wait_asynccnt)
#define HAVE_WAIT_ASYNC_BUILTIN 1
#endif
#endif

__device__ __forceinline__ void async_copy16(const float* g, float* l) {
#ifdef HAVE_ASYNC_LDS_BUILTIN
  __builtin_amdgcn_global_load_async_to_lds_b128(
      (__attribute__((address_space(1))) v4i*)g,
      (__attribute__((address_space(3))) v4i*)l, 0, 0);
#else
  // VFLAT-style operand order (dst, vaddr, off) per gfx1250 disassembly.
  asm volatile("global_load_async_to_lds_b128 %0, %1, off"
               :: "v"((unsigned)(size_t)l), "v"(g) : "memory");
#endif
}

__device__ __forceinline__ void wait_async0() {
#ifdef HAVE_WAIT_ASYNC_BUILTIN
  __builtin_amdgcn_s_wait_asynccnt(0);
#else
  asm volatile("s_wait_asynccnt 0x0" ::: "memory");
#endif
}

__device__ __forceinline__ v8f wmma_f32_16x16x4(v2f a, v2f b, v8f c) {
  // 8 args: (neg_a, A, neg_b, B, c_mod, C, reuse_a, reuse_b)
  return __builtin_amdgcn_wmma_f32_16x16x4_f32(false, a, false, b, (short)0, c,
                                               false, false);
}

__global__ __launch_bounds__(256)
void gnn_local_cluster_fused(const float* __restrict__ x_in,
                             const float* __restrict__ Wf,
                             const float* __restrict__ bf,
                             const float* __restrict__ Wp,
                             const float* __restrict__ bp,
                             const float* __restrict__ W1,
                             const float* __restrict__ b1,
                             const float* __restrict__ W2,
                             const float* __restrict__ b2,
                             const float* __restrict__ sigma_p,
                             const float* __restrict__ alpha_p,
                             float* __restrict__ y)
{
  extern __shared__ char smem_raw[];
  float* xs = (float*)smem_raw;                         // 64*256 f32 = 64KB (reused)
  float* xf = (float*)(smem_raw + 65536);               // 256*17 f32
  float* xn = xf + NN * XF_STR;                         // 256*17 f32
  float* agg = xn + NN * XF_STR;                        // 256*17 f32
  // overlay region inside xs after phase A:
  float* sim      = xs;                                 // 16*273 f32
  float* cand_val = sim + 16 * SIM_STR;                 // 16*16*9 f32
  int*   cand_idx = (int*)(cand_val + 16 * 16 * KSEL);  // 16*16*9 i32
  float* wk_w     = (float*)(cand_idx + 16 * 16 * KSEL);// 16*9 f32
  int*   wk_i     = (int*)(wk_w + 16 * KSEL);           // 16*9 i32

  const int t    = threadIdx.x;
  const int lane = t & 31;
  const int wv   = t >> 5;          // wave id 0..7
  const int half = (lane >> 4) & 1; // lane half (K split for 32-bit WMMA frags)
  const int nl   = lane & 15;       // within-half lane: column / row-M index

  const int b  = blockIdx.x;        // 0..783
  const int Bo = b / 49;
  const int sw = b % 49;
  const int s1 = sw / 7;
  const int s2 = sw % 7;

  const float* xbase = x_in + (size_t)Bo * C_IN * HW + (s1 * 16) * 112 + (s2 * 16);

  // ---- Phase A1: async-stage x window tile into LDS: xs[c*256 + n] ----
  // 4096 16-byte chunks (chunk = c*64 + i*4 + j4), 16 per thread; each chunk
  // is 16B-aligned in both global and LDS. Overlaps with Wf fragment loads.
  {
#pragma unroll
    for (int k = 0; k < 16; ++k) {
      const int chunk = t + 256 * k;
      const int c   = chunk >> 6;
      const int rem = chunk & 63;
      const int i   = rem >> 2;
      const int j4  = rem & 3;
      async_copy16(xbase + c * HW + i * 112 + j4 * 4,
                   xs + c * NN + i * 16 + j4 * 4);
    }
  }

  // ---- Phase A0: preload Wf A-fragments (16x64, K in 16 steps of 4) ----
  v2f afr[16];
#pragma unroll
  for (int kk = 0; kk < 16; ++kk) {
    const int kb = kk * 4 + 2 * half;
    afr[kk].x = Wf[nl * 64 + kb];
    afr[kk].y = Wf[nl * 64 + kb + 1];
  }

  wait_async0();
  __syncthreads();

  // ---- Phase A2: xf = Wf * x + bf  (WMMA, 2 node-tiles per wave) ----
#pragma unroll
  for (int tt = 0; tt < 2; ++tt) {
    const int nt = wv * 2 + tt;
    v8f acc;
#pragma unroll
    for (int vg = 0; vg < 8; ++vg) acc[vg] = bf[vg + 8 * half];
#pragma unroll
    for (int kk = 0; kk < 16; ++kk) {
      const int kb = kk * 4 + 2 * half;
      v2f bfr;
      bfr.x = xs[(kb + 0) * NN + nt * 16 + nl];
      bfr.y = xs[(kb + 1) * NN + nt * 16 + nl];
      acc = wmma_f32_16x16x4(afr[kk], bfr, acc);
    }
#pragma unroll
    for (int vg = 0; vg < 8; ++vg)
      xf[(nt * 16 + nl) * XF_STR + (vg + 8 * half)] = acc[vg];
  }
  __syncthreads();

  // ---- Phase B: L2-normalize feature rows ----
  {
    const int n = t;
    float ss = 0.f;
#pragma unroll
    for (int c = 0; c < CF; ++c) { float v = xf[n * XF_STR + c]; ss += v * v; }
    const float rn = 1.0f / fmaxf(sqrtf(ss), 1e-8f);
#pragma unroll
    for (int c = 0; c < CF; ++c) xn[n * XF_STR + c] = xf[n * XF_STR + c] * rn;
  }
  __syncthreads();

  const float alpha = alpha_p[0];
  const float sg = sigma_p[0];
  const float inv2s2 = 1.0f / (2.0f * sg * sg);
  const float oma = 1.0f - alpha;

  // ---- Phase C/D: per 16-row tile: WMMA sim block, top-9, MLP, aggregate ----
  for (int rt = 0; rt < 16; ++rt) {
    // C: sim_feat rows [rt*16, rt*16+16) x all 256 cols, 2 col-tiles per wave
#pragma unroll
    for (int tt = 0; tt < 2; ++tt) {
      const int ct = wv * 2 + tt;
      v8f acc = {};
#pragma unroll
      for (int kk = 0; kk < 4; ++kk) {
        const int kb = kk * 4 + 2 * half;
        v2f a, bb;
        a.x  = xn[(rt * 16 + nl) * XF_STR + kb];
        a.y  = xn[(rt * 16 + nl) * XF_STR + kb + 1];
        bb.x = xn[(ct * 16 + nl) * XF_STR + kb];
        bb.y = xn[(ct * 16 + nl) * XF_STR + kb + 1];
        acc = wmma_f32_16x16x4(a, bb, acc);
      }
#pragma unroll
      for (int vg = 0; vg < 8; ++vg)
        sim[(vg + 8 * half) * SIM_STR + ct * 16 + nl] = acc[vg];
    }
    __syncthreads();

    // D1: strip-parallel top-9 of combined similarity (16 strips per node)
    {
      const int r  = t & 15;
      const int sp = t >> 4;
      const int n  = rt * 16 + r;
      const int ni = n >> 4, nj = n & 15;
      float v9[KSEL]; int i9[KSEL];
#pragma unroll
      for (int q = 0; q < KSEL; ++q) { v9[q] = -1e30f; i9[q] = 0; }
#pragma unroll
      for (int mm = 0; mm < 16; ++mm) {
        const int m  = sp * 16 + mm;
        const int mi = m >> 4, mj = m & 15;
        const int di = ni - mi, dj = nj - mj;
        const float sd = __expf(-(float)(di * di + dj * dj) * inv2s2);
        const float cv = alpha * sim[r * SIM_STR + m] + oma * sd;
        if (cv > v9[KSEL - 1]) {
          v9[KSEL - 1] = cv; i9[KSEL - 1] = m;
#pragma unroll
          for (int q = KSEL - 1; q > 0; --q) {
            if (v9[q] > v9[q - 1]) {
              float tv = v9[q]; v9[q] = v9[q - 1]; v9[q - 1] = tv;
              int   ti = i9[q]; i9[q] = i9[q - 1]; i9[q - 1] = ti;
            }
          }
        }
      }
#pragma unroll
      for (int q = 0; q < KSEL; ++q) {
        cand_val[(r * 16 + sp) * KSEL + q] = v9[q];
        cand_idx[(r * 16 + sp) * KSEL + q] = i9[q];
      }
    }
    __syncthreads();

    // D2: merge 144 candidates -> top-9, run edge MLP, normalize weights
    if (t < 16) {
      const int r = t;
      const int n = rt * 16 + r;
      const int ni = n >> 4, nj = n & 15;
      float v9[KSEL]; int i9[KSEL];
#pragma unroll
      for (int q = 0; q < KSEL; ++q) { v9[q] = -1e30f; i9[q] = 0; }
      for (int e = 0; e < 16 * KSEL; ++e) {
        const float cv = cand_val[r * 16 * KSEL + e];
        const int   ci = cand_idx[r * 16 * KSEL + e];
        if (cv > v9[KSEL - 1]) {
          v9[KSEL - 1] = cv; i9[KSEL - 1] = ci;
#pragma unroll
          for (int q = KSEL - 1; q > 0; --q) {
            if (v9[q] > v9[q - 1]) {
              float tv = v9[q]; v9[q] = v9[q - 1]; v9[q - 1] = tv;
              int   ti = i9[q]; i9[q] = i9[q - 1]; i9[q - 1] = ti;
            }
          }
        }
      }
      float wsum = 0.f, wq[KSEL];
#pragma unroll
      for (int q = 0; q < KSEL; ++q) {
        const int m  = i9[q];
        const int mi = m >> 4, mj = m & 15;
        const int di = ni - mi, dj = nj - mj;
        const float sd = __expf(-(float)(di * di + dj * dj) * inv2s2);
        const float sf = sim[r * SIM_STR + m];
        float hsum = b2[0];
#pragma unroll
        for (int j4 = 0; j4 < 4; ++j4) {
          float hv = sf * W1[j4 * 2 + 0] + sd * W1[j4 * 2 + 1] + b1[j4];
          hv = hv / (1.0f + __expf(-hv));          // SiLU
          hsum += hv * W2[j4];
        }
        const float wgt = 1.0f / (1.0f + __expf(-hsum));  // sigmoid
        wq[q] = wgt; wsum += wgt;
      }
      const float rs = 1.0f / (wsum + 1e-12f);
#pragma unroll
      for (int q = 0; q < KSEL; ++q) {
        wk_w[r * KSEL + q] = wq[q] * rs;
        wk_i[r * KSEL + q] = i9[q];
      }
    }
    __syncthreads();

    // D3: weighted message aggregation (thread = (node r, channel c))
    {
      const int r = t & 15;
      const int c = t >> 4;
      const int n = rt * 16 + r;
      float a = 0.f;
#pragma unroll
      for (int q = 0; q < KSEL; ++q)
        a += wk_w[r * KSEL + q] * xf[wk_i[r * KSEL + q] * XF_STR + c];
      agg[n * XF_STR + c] = a;
    }
    __syncthreads();
  }

  // ---- Phase E: y = Wp * agg^T + bp  (WMMA, 8 tiles per wave) ----
  float* ybase = y + (size_t)Bo * C_IN * HW + (s1 * 16) * 112 + (s2 * 16);
#pragma unroll
  for (int tt = 0; tt < 8; ++tt) {
    const int tile = wv * 8 + tt;     // 0..63
    const int mt = tile >> 4;         // out-channel tile 0..3
    const int nt = tile & 15;         // node tile 0..15 (fixed grid row i=nt)
    v8f acc;
#pragma unroll
    for (int vg = 0; vg < 8; ++vg) acc[vg] = bp[mt * 16 + vg + 8 * half];
#pragma unroll
    for (int kk = 0; kk < 4; ++kk) {
      const int kb = kk * 4 + 2 * half;
      v2f a, bb;
      a.x  = Wp[(mt * 16 + nl) * CF + kb];
      a.y  = Wp[(mt * 16 + nl) * CF + kb + 1];
      bb.x = agg[(nt * 16 + nl) * XF_STR + kb];
      bb.y = agg[(nt * 16 + nl) * XF_STR + kb + 1];
      acc = wmma_f32_16x16x4(a, bb, acc);
    }
#pragma unroll
    for (int vg = 0; vg < 8; ++vg) {
      const int o = mt * 16 + vg + 8 * half;
      ybase[o * HW + nt * 112 + nl] = acc[vg];   // 64B coalesced per half-wave
    }
  }
}

extern "C" void kernel_launch(void* const* d_in, const int* in_sizes, int n_in,
                              void* d_out, int out_size, void* d_ws, size_t ws_size,
                              hipStream_t stream) {
  (void)in_sizes; (void)n_in; (void)out_size; (void)d_ws; (void)ws_size;
  const float* x_in  = (const float*)d_in[0];
  const float* Wf    = (const float*)d_in[1];
  const float* bf    = (const float*)d_in[2];
  const float* Wp    = (const float*)d_in[3];
  const float* bp    = (const float*)d_in[4];
  const float* W1    = (const float*)d_in[5];
  const float* b1    = (const float*)d_in[6];
  const float* W2    = (const float*)d_in[7];
  const float* b2    = (const float*)d_in[8];
  const float* sigma = (const float*)d_in[9];
  const float* alpha = (const float*)d_in[10];
  float* yout = (float*)d_out;

  // dynamic LDS: 64KB staging/overlay + 3 padded 256x17 f32 planes = 115KB
  const int smem = 65536 + 3 * (NN * XF_STR * (int)sizeof(float));
  gnn_local_cluster_fused<<<784, 256, smem, stream>>>(
      x_in, Wf, bf, Wp, bp, W1, b1, W2, b2, sigma, alpha, yout);
}